// Rescaler_45810121179193
// MI455X (gfx1250) — compile-verified
//
#include <hip/hip_runtime.h>
#include <hip/hip_bf16.h>

#define BINS        256
#define B_SAMPLES   128
#define HW          262144          // 512*512 elements per sample (C=1)
#define CHUNKS      32              // histogram blocks per sample
#define CHUNK_ELEMS (HW / CHUNKS)   // 8192

typedef float v2f __attribute__((ext_vector_type(2)));
typedef float v8f __attribute__((ext_vector_type(8)));

__device__ __forceinline__ v8f wmma_f32_16x16x4(v2f a, v2f b, v8f c) {
    // 8 args: (neg_a, A, neg_b, B, c_mod, C, reuse_a, reuse_b)
    return __builtin_amdgcn_wmma_f32_16x16x4_f32(
        false, a, false, b, (short)0, c, false, false);
}

// ---------------------------------------------------------------------------
// 1) zero the per-sample histogram accumulators (atomics -> must zero per call)
// ---------------------------------------------------------------------------
__global__ void zero_hist_kernel(int* __restrict__ hstI) {
    hstI[blockIdx.x * 256 + threadIdx.x] = 0;
}

// ---------------------------------------------------------------------------
// 2) per-sample histogram via LDS atomics; float4 loads for bandwidth
// ---------------------------------------------------------------------------
__global__ void __launch_bounds__(256)
hist_kernel(const float* __restrict__ x, int* __restrict__ hstI) {
    __shared__ int lh[BINS];
    const int t = threadIdx.x;
    lh[t] = 0;
    __syncthreads();

    const int b = blockIdx.y;
    const float4* xs = (const float4*)(x + (size_t)b * HW +
                                       (size_t)blockIdx.x * CHUNK_ELEMS);
    #pragma unroll
    for (int it = 0; it < CHUNK_ELEMS / 4 / 256; ++it) {   // 8 iterations
        float4 v = xs[it * 256 + t];
        float vv[4] = {v.x, v.y, v.z, v.w};
        #pragma unroll
        for (int j = 0; j < 4; ++j) {
            float f = vv[j];
            if (f >= 0.0f && f <= 1.0f) {                  // histc: ignore OOR
                int idx = (int)(f * 256.0f);               // trunc==floor, f>=0
                idx = idx > 255 ? 255 : idx;               // f==1.0 -> last bin
                atomicAdd(&lh[idx], 1);
            }
        }
    }
    __syncthreads();
    const int c = lh[t];
    if (c) atomicAdd(&hstI[b * BINS + t], c);
}

// ---------------------------------------------------------------------------
// 3) threshold value per sample (first-max argmax, then first bin <= max*rate)
// ---------------------------------------------------------------------------
__global__ void thresh_kernel(const int* __restrict__ hstI,
                              float* __restrict__ valueOut) {
    const int b = threadIdx.x;
    if (b >= B_SAMPLES) return;
    const int* h = hstI + b * BINS;
    int maxv = h[0], amax = 0;
    for (int i = 1; i < BINS; ++i) {
        int v = h[i];
        if (v > maxv) { maxv = v; amax = i; }              // first occurrence
    }
    const float height = (float)maxv * 0.05f;
    int res = 0;                                           // argmin default
    for (int i = amax; i < BINS; ++i) {
        if ((float)h[i] <= height) { res = i; break; }
    }
    valueOut[b] = (float)res / (float)BINS;
}

// ---------------------------------------------------------------------------
// 4) int histogram -> float
// ---------------------------------------------------------------------------
__global__ void cvt_hist_kernel(const int* __restrict__ hstI,
                                float* __restrict__ hstF) {
    int i = blockIdx.x * 256 + threadIdx.x;
    hstF[i] = (float)hstI[i];
}

// ---------------------------------------------------------------------------
// 5) MLP via f32 WMMA 16x16x4.  H = relu(X @ W^T + bias)
//    A lane layout: M=l&15, K=v+2*(l>>4);  B: N=l&15, K=v+2*(l>>4)
//    D lane layout: M=v+8*(l>>4), N=l&15
// ---------------------------------------------------------------------------
template <int K, int N, int TPW>
__device__ __forceinline__ void mlp_layer(const float* __restrict__ X,
                                          const float* __restrict__ Wt,
                                          const float* __restrict__ bias,
                                          float* __restrict__ H,
                                          int wave, int lane) {
    const int row  = lane & 15;
    const int koff = (lane >> 4) << 1;
    for (int i = 0; i < TPW; ++i) {
        const int ti = wave * TPW + i;
        const int m0 = (ti & 7) * 16;        // 8 row-tiles (M=128)
        const int n0 = (ti >> 3) * 16;       // N/16 col-tiles
        v8f acc = {};
        for (int k0 = 0; k0 < K; k0 += 4) {
            v2f a = *(const v2f*)(X  + (m0 + row) * K + k0 + koff);
            v2f b = *(const v2f*)(Wt + (n0 + row) * K + k0 + koff);
            acc = wmma_f32_16x16x4(a, b, acc);
        }
        const int n  = n0 + row;
        const float bn = bias[n];
        #pragma unroll
        for (int v = 0; v < 8; ++v) {
            const int m = m0 + v + ((lane >> 4) << 3);
            float val = acc[v] + bn;
            H[m * N + n] = val > 0.0f ? val : 0.0f;
        }
    }
}

__global__ void __launch_bounds__(256)
mlp_kernel(const float* __restrict__ hstF,
           const float* __restrict__ W1, const float* __restrict__ b1,
           const float* __restrict__ W2, const float* __restrict__ b2,
           const float* __restrict__ W3, const float* __restrict__ b3,
           const float* __restrict__ W4, const float* __restrict__ b4,
           float* __restrict__ h3ws, float* __restrict__ wOut) {
    __shared__ float h1[B_SAMPLES * 32];     // 16 KiB
    __shared__ float h2[B_SAMPLES * 64];     // 32 KiB
    const int lane = threadIdx.x & 31;
    const int wave = threadIdx.x >> 5;

    mlp_layer<256, 32, 2>(hstF, W1, b1, h1, wave, lane);   // 16 tiles / 8 waves
    __syncthreads();
    mlp_layer<32, 64, 4>(h1, W2, b2, h2, wave, lane);      // 32 tiles
    __syncthreads();
    mlp_layer<64, 128, 8>(h2, W3, b3, h3ws, wave, lane);   // 64 tiles -> ws
    __threadfence();
    __syncthreads();

    if (threadIdx.x < B_SAMPLES) {           // final 128->1 dot per sample
        float s = b4[0];
        const float* hr = h3ws + threadIdx.x * 128;
        #pragma unroll 8
        for (int k = 0; k < 128; ++k) s += hr[k] * W4[k];
        wOut[threadIdx.x] = s;
    }
}

// ---------------------------------------------------------------------------
// 6) out = x * w[b]   (float4 vectorized; x should be L2-resident, 192 MB L2)
// ---------------------------------------------------------------------------
__global__ void __launch_bounds__(256)
scale_kernel(const float4* __restrict__ x4, const float* __restrict__ w,
             float4* __restrict__ out4) {
    const unsigned i = blockIdx.x * 256u + threadIdx.x;    // 8388608 float4s
    const float s = w[i >> 16];                            // 65536 float4/sample
    float4 v = x4[i];
    v.x *= s; v.y *= s; v.z *= s; v.w *= s;
    out4[i] = v;
}

// ---------------------------------------------------------------------------
extern "C" void kernel_launch(void* const* d_in, const int* in_sizes, int n_in,
                              void* d_out, int out_size, void* d_ws, size_t ws_size,
                              hipStream_t stream) {
    const float* x  = (const float*)d_in[0];
    const float* W1 = (const float*)d_in[1];
    const float* b1 = (const float*)d_in[2];
    const float* W2 = (const float*)d_in[3];
    const float* b2 = (const float*)d_in[4];
    const float* W3 = (const float*)d_in[5];
    const float* b3 = (const float*)d_in[6];
    const float* W4 = (const float*)d_in[7];
    const float* b4 = (const float*)d_in[8];

    char*  ws    = (char*)d_ws;
    int*   hstI  = (int*)ws;                       // 128*256 ints   (128 KiB)
    float* hstF  = (float*)(ws + 131072);          // 128*256 floats (128 KiB)
    float* h3    = (float*)(ws + 262144);          // 128*128 floats ( 64 KiB)
    float* wOut  = (float*)(ws + 327680);          // 128 floats

    float* out   = (float*)d_out;
    float* value = out + (size_t)B_SAMPLES * HW;   // value tail of d_out

    zero_hist_kernel<<<128, 256, 0, stream>>>(hstI);
    hist_kernel<<<dim3(CHUNKS, B_SAMPLES), 256, 0, stream>>>(x, hstI);
    thresh_kernel<<<1, 128, 0, stream>>>(hstI, value);
    cvt_hist_kernel<<<128, 256, 0, stream>>>(hstI, hstF);
    mlp_kernel<<<1, 256, 0, stream>>>(hstF, W1, b1, W2, b2, W3, b3, W4, b4,
                                      h3, wOut);
    scale_kernel<<<HW * B_SAMPLES / 4 / 256, 256, 0, stream>>>(
        (const float4*)x, wOut, (float4*)out);
}